// GNNScentClassifier_66279935312387
// MI455X (gfx1250) — compile-verified
//
#include <hip/hip_runtime.h>
#include <hip/hip_bf16.h>
#include <math.h>

#define N_NODES  100000
#define N_EDGES  1600000
#define N_GRAPHS 2048
#define HH       43
#define HP       48
#define H3P      144
#define FF       392
#define FFP      400
#define NLB      138
#define NLP      144
#define N_LAYERS 5

typedef __attribute__((ext_vector_type(2))) float v2f;
typedef __attribute__((ext_vector_type(8))) float v8f;

// ---------------- WMMA helper (V_WMMA_F32_16X16X4_F32, wave32) ----------------

__device__ __forceinline__ v8f wmma4(v2f a, v2f b, v8f c) {
  // 8 args: (neg_a, A, neg_b, B, c_mod, C, reuse_a, reuse_b)
  return __builtin_amdgcn_wmma_f32_16x16x4_f32(false, a, false, b, (short)0, c,
                                               false, false);
}

__device__ __forceinline__ float sigmoidf_(float x) {
  return 1.0f / (1.0f + __expf(-x));
}

// Packed-B layout: K-pair interleaved. Element B[k][n] lives at
//   Bp[ ((k>>1)*Nb + n)*2 + (k&1) ]
// so a B fragment (lanes 0-15: rows {k0,k0+1}; lanes 16-31: rows {k0+2,k0+3})
// is ONE 64-bit load per lane: pair row (k0>>1)+(lane>>4), column n0+(lane&15).

// ---------------- elementwise / packing kernels ----------------

__global__ void zero_f32(float* __restrict__ p, unsigned n) {
  unsigned i = blockIdx.x * blockDim.x + threadIdx.x;
  if (i < n) p[i] = 0.0f;
}

__global__ void pad_x_kernel(const float* __restrict__ x, float* __restrict__ h) {
  unsigned i = blockIdx.x * blockDim.x + threadIdx.x;
  if (i >= (unsigned)N_NODES * HP) return;
  unsigned n = i / HP, c = i - n * HP;
  h[i] = (c < HH) ? x[(size_t)n * HH + c] : 0.0f;
}

// W[5][43][43] -> per-layer pair-interleaved [48/2][48][2]
__global__ void pack_wl(const float* __restrict__ W, float* __restrict__ out) {
  unsigned i = blockIdx.x * blockDim.x + threadIdx.x;
  if (i >= N_LAYERS * HP * HP) return;
  unsigned l = i / (HP * HP), r = i % (HP * HP), k = r / HP, n = r % HP;
  float v = (k < HH && n < HH) ? W[(size_t)l * HH * HH + (size_t)k * HH + n] : 0.0f;
  out[(size_t)l * HP * HP + (((k >> 1) * HP + n) << 1) + (k & 1)] = v;
}

// W_ih[129][43] -> B = W_ih^T grouped (col = g*48+j), pair-interleaved [48/2][144][2]
__global__ void pack_gru_w(const float* __restrict__ Wg, float* __restrict__ out) {
  unsigned i = blockIdx.x * blockDim.x + threadIdx.x;
  if (i >= HP * H3P) return;
  unsigned k = i / H3P, col = i % H3P, g = col / HP, j = col % HP;
  float v = (k < HH && j < HH) ? Wg[(size_t)(g * HH + j) * HH + k] : 0.0f;
  out[(((k >> 1) * H3P + col) << 1) + (k & 1)] = v;
}

// Win[Nv][Kv] -> B[k][n] = Win[n][k], pair-interleaved [Kp/2][Np][2]
__global__ void pack_t(const float* __restrict__ Win, float* __restrict__ out,
                       int Kv, int Nv, int Kp, int Np) {
  unsigned i = blockIdx.x * blockDim.x + threadIdx.x;
  if (i >= (unsigned)(Kp * Np)) return;
  unsigned k = i / Np, n = i % Np;
  float v = ((int)k < Kv && (int)n < Nv) ? Win[(size_t)n * Kv + k] : 0.0f;
  out[(((k >> 1) * Np + n) << 1) + (k & 1)] = v;
}

// ---------------- edge scatter-add: agg[dst] += m[src] ----------------

__global__ void scatter_add(const int* __restrict__ ei, const float* __restrict__ m,
                            float* __restrict__ agg) {
  unsigned i = blockIdx.x * blockDim.x + threadIdx.x;
  if (i >= (unsigned)N_EDGES * HP) return;   // 76.8M < 2^31
  unsigned e = i / HP, c = i - e * HP;
  if (c >= HH) return;
  int src = ei[e];
  int dst = ei[N_EDGES + e];
  unsafeAtomicAdd(agg + (size_t)dst * HP + c, m[(size_t)src * HP + c]);
}

// ---------------- global_add_pool via atomics ----------------

__global__ void pool_add(const float* __restrict__ h, const int* __restrict__ batch,
                         float* __restrict__ g) {
  unsigned i = blockIdx.x * blockDim.x + threadIdx.x;
  if (i >= (unsigned)N_NODES * HP) return;
  unsigned n = i / HP, c = i - n * HP;
  if (c >= HH) return;
  unsafeAtomicAdd(g + (size_t)batch[n] * HP + c, h[(size_t)n * HP + c]);
}

// ---------------- generic WMMA f32 GEMM: C = act(A[M,K] @ B[K,Nb] + bias) ------------
// B is pair-interleaved (see above). Inner loop: 2 loads + 2 pointer bumps + 1 WMMA.

__global__ void wmma_gemm(const float* __restrict__ A, const float* __restrict__ Bp,
                          const float* __restrict__ bias, int bias_n,
                          float* __restrict__ C, int M, int K, int Nb,
                          int ldc, int out_n, int act) {
  int lane = threadIdx.x & 31;
  int wid  = (int)((blockIdx.x * blockDim.x + threadIdx.x) >> 5);
  int tiles_n = Nb >> 4, tiles_m = M >> 4;
  if (wid >= tiles_m * tiles_n) return;           // wave-uniform guard
  int tm = wid / tiles_n, tn = wid - tm * tiles_n;
  int rb = tm << 4, n0 = tn << 4;
  int half = lane >> 4, lm = lane & 15;

  const float*  pa = A + (size_t)(rb + lm) * K + (half << 1);
  const float2* pb = (const float2*)Bp + (size_t)half * Nb + n0 + lm;
  const size_t  bstep = 2u * (size_t)Nb;          // advance 2 pair-rows per k-step

  v8f acc = {};
  for (int k0 = 0; k0 < K; k0 += 4) {
    v2f a{pa[0], pa[1]};                          // global_load_b64
    float2 t = *pb;                               // global_load_b64
    v2f b{t.x, t.y};
    acc = wmma4(a, b, acc);
    pa += 4;
    pb += bstep;
  }

  int col = n0 + lm;
  float bv = (bias != nullptr && col < bias_n) ? bias[col] : 0.0f;
  if (col < out_n) {
#pragma unroll
    for (int v = 0; v < 8; ++v) {
      float xv = acc[v] + bv;
      if (act == 1) xv = fmaxf(xv, 0.0f);
      else if (act == 2) xv = sigmoidf_(xv);
      C[(size_t)(rb + v + 8 * half) * ldc + col] = xv;
    }
  }
}

// ---------------- fused GRU cell: 6 WMMA chains + gates per 16x16 tile ----------------
// gi = agg @ Bih (+b_ih), gh = hold @ Bhh (+b_hh); r,z,n gates; hnew = (1-z)*n + z*hold

__global__ void gru_fused(const float* __restrict__ agg, const float* __restrict__ hold,
                          const float* __restrict__ Bih, const float* __restrict__ Bhh,
                          const float* __restrict__ bih, const float* __restrict__ bhh,
                          float* __restrict__ hnew) {
  int lane = threadIdx.x & 31;
  int wid  = (int)((blockIdx.x * blockDim.x + threadIdx.x) >> 5);
  const int tiles_m = N_NODES / 16;                // 6250
  if (wid >= tiles_m * 3) return;                  // wave-uniform guard
  int tm = wid / 3, ct = wid - tm * 3;             // ct: output column tile (0..2)
  int rb = tm << 4;
  int half = lane >> 4, lm = lane & 15;

  const float* paa = agg  + (size_t)(rb + lm) * HP + (half << 1);
  const float* pah = hold + (size_t)(rb + lm) * HP + (half << 1);
  const float2* base_i = (const float2*)Bih + (size_t)half * H3P + lm + (ct << 4);
  const float2* base_h = (const float2*)Bhh + (size_t)half * H3P + lm + (ct << 4);
  const size_t bstep = 2u * (size_t)H3P;

  const float2* pbr_i = base_i;            // r gate cols [0*48 ..)
  const float2* pbz_i = base_i + HP;       // z gate cols [1*48 ..)
  const float2* pbn_i = base_i + 2 * HP;   // n gate cols [2*48 ..)
  const float2* pbr_h = base_h;
  const float2* pbz_h = base_h + HP;
  const float2* pbn_h = base_h + 2 * HP;

  v8f ir = {}, iz = {}, in_ = {}, hr = {}, hz = {}, hn = {};
  for (int k0 = 0; k0 < HP; k0 += 4) {
    v2f aa{paa[0], paa[1]};
    v2f ah{pah[0], pah[1]};
    float2 t;
    t = *pbr_i; ir  = wmma4(aa, v2f{t.x, t.y}, ir);
    t = *pbr_h; hr  = wmma4(ah, v2f{t.x, t.y}, hr);
    t = *pbz_i; iz  = wmma4(aa, v2f{t.x, t.y}, iz);
    t = *pbz_h; hz  = wmma4(ah, v2f{t.x, t.y}, hz);
    t = *pbn_i; in_ = wmma4(aa, v2f{t.x, t.y}, in_);
    t = *pbn_h; hn  = wmma4(ah, v2f{t.x, t.y}, hn);
    paa += 4; pah += 4;
    pbr_i += bstep; pbz_i += bstep; pbn_i += bstep;
    pbr_h += bstep; pbz_h += bstep; pbn_h += bstep;
  }

  int j = (ct << 4) + lm;
  bool valid = j < HH;
  float bir = valid ? bih[j] : 0.0f;
  float biz = valid ? bih[HH + j] : 0.0f;
  float bin = valid ? bih[2 * HH + j] : 0.0f;
  float bhr = valid ? bhh[j] : 0.0f;
  float bhz = valid ? bhh[HH + j] : 0.0f;
  float bhn = valid ? bhh[2 * HH + j] : 0.0f;

#pragma unroll
  for (int v = 0; v < 8; ++v) {
    int row = rb + v + 8 * half;
    float r = sigmoidf_(ir[v] + bir + hr[v] + bhr);
    float z = sigmoidf_(iz[v] + biz + hz[v] + bhz);
    float n = tanhf(in_[v] + bin + r * (hn[v] + bhn));
    float hp = hold[(size_t)row * HP + j];
    hnew[(size_t)row * HP + j] = (1.0f - z) * n + z * hp;   // padded cols stay 0
  }
}

// ---------------- host-side orchestration ----------------

static inline int cdiv(long long a, long long b) { return (int)((a + b - 1) / b); }

extern "C" void kernel_launch(void* const* d_in, const int* in_sizes, int n_in,
                              void* d_out, int out_size, void* d_ws, size_t ws_size,
                              hipStream_t stream) {
  const float* x    = (const float*)d_in[0];
  const int*   ei   = (const int*)d_in[1];
  const int*   bat  = (const int*)d_in[2];
  const float* W    = (const float*)d_in[3];
  const float* W_ih = (const float*)d_in[4];
  const float* W_hh = (const float*)d_in[5];
  const float* b_ih = (const float*)d_in[6];
  const float* b_hh = (const float*)d_in[7];
  const float* W1   = (const float*)d_in[8];
  const float* b1   = (const float*)d_in[9];
  const float* W2   = (const float*)d_in[10];
  const float* b2   = (const float*)d_in[11];
  const float* W3   = (const float*)d_in[12];
  const float* b3   = (const float*)d_in[13];
  const float* Wp   = (const float*)d_in[14];
  const float* bp   = (const float*)d_in[15];
  float* out = (float*)d_out;
  (void)in_sizes; (void)n_in; (void)out_size; (void)ws_size;

  // workspace layout (floats)
  float* ws   = (float*)d_ws;
  float* hA   = ws;
  float* hB   = hA   + (size_t)N_NODES * HP;
  float* mbuf = hB   + (size_t)N_NODES * HP;
  float* agg  = mbuf + (size_t)N_NODES * HP;
  float* Wl   = agg  + (size_t)N_NODES * HP;       // 5*48*48
  float* Bih  = Wl   + (size_t)N_LAYERS * HP * HP; // 48*144
  float* Bhh  = Bih  + (size_t)HP * H3P;
  float* B1   = Bhh  + (size_t)HP * H3P;           // 48*400
  float* B2   = B1   + (size_t)HP * FFP;           // 400*400
  float* B3   = B2   + (size_t)FFP * FFP;
  float* Bp   = B3   + (size_t)FFP * FFP;          // 400*144
  float* gpool= Bp   + (size_t)FFP * NLP;          // 2048*48
  float* f1   = gpool+ (size_t)N_GRAPHS * HP;      // 2048*400
  float* f2   = f1   + (size_t)N_GRAPHS * FFP;

  const int EB = 256;

  // input padding + weight packing (pair-interleaved B layouts)
  pad_x_kernel<<<cdiv((long long)N_NODES * HP, EB), EB, 0, stream>>>(x, hA);
  pack_wl   <<<cdiv(N_LAYERS * HP * HP, EB), EB, 0, stream>>>(W, Wl);
  pack_gru_w<<<cdiv(HP * H3P, EB), EB, 0, stream>>>(W_ih, Bih);
  pack_gru_w<<<cdiv(HP * H3P, EB), EB, 0, stream>>>(W_hh, Bhh);
  pack_t    <<<cdiv(HP * FFP, EB), EB, 0, stream>>>(W1, B1, HH, FF, HP, FFP);
  pack_t    <<<cdiv(FFP * FFP, EB), EB, 0, stream>>>(W2, B2, FF, FF, FFP, FFP);
  pack_t    <<<cdiv(FFP * FFP, EB), EB, 0, stream>>>(W3, B3, FF, FF, FFP, FFP);
  pack_t    <<<cdiv(FFP * NLP, EB), EB, 0, stream>>>(Wp, Bp, FF, NLB, FFP, NLP);

  // 5 GatedGraphConv layers (ping-pong hA <-> hB)
  const int gemm_waves_node = (N_NODES / 16) * (HP / 16);   // 18750
  const int gru_waves       = (N_NODES / 16) * 3;           // 18750
  for (int l = 0; l < N_LAYERS; ++l) {
    float* hin  = (l & 1) ? hB : hA;
    float* hout = (l & 1) ? hA : hB;
    // m = h @ W[l]
    wmma_gemm<<<cdiv((long long)gemm_waves_node * 32, 128), 128, 0, stream>>>(
        hin, Wl + (size_t)l * HP * HP, nullptr, 0, mbuf,
        N_NODES, HP, HP, HP, HP, 0 /*none*/);
    // agg = 0 ; agg[dst] += m[src]
    zero_f32<<<cdiv((long long)N_NODES * HP, EB), EB, 0, stream>>>(agg, (unsigned)(N_NODES * HP));
    scatter_add<<<cdiv((long long)N_EDGES * HP, EB), EB, 0, stream>>>(ei, mbuf, agg);
    // GRU cell
    gru_fused<<<cdiv((long long)gru_waves * 32, 128), 128, 0, stream>>>(
        agg, hin, Bih, Bhh, b_ih, b_hh, hout);
  }
  float* hfin = hB;   // after 5 layers: A->B,B->A,A->B,B->A,A->B

  // global_add_pool
  zero_f32<<<cdiv((long long)N_GRAPHS * HP, EB), EB, 0, stream>>>(gpool, (unsigned)(N_GRAPHS * HP));
  pool_add<<<cdiv((long long)N_NODES * HP, EB), EB, 0, stream>>>(hfin, bat, gpool);

  // MLP head
  int w1 = (N_GRAPHS / 16) * (FFP / 16);   // 3200 waves
  int wp = (N_GRAPHS / 16) * (NLP / 16);   // 1152 waves
  wmma_gemm<<<cdiv((long long)w1 * 32, 128), 128, 0, stream>>>(
      gpool, B1, b1, FF, f1, N_GRAPHS, HP, FFP, FFP, FFP, 1 /*relu*/);
  wmma_gemm<<<cdiv((long long)w1 * 32, 128), 128, 0, stream>>>(
      f1, B2, b2, FF, f2, N_GRAPHS, FFP, FFP, FFP, FFP, 1 /*relu*/);
  wmma_gemm<<<cdiv((long long)w1 * 32, 128), 128, 0, stream>>>(
      f2, B3, b3, FF, f1, N_GRAPHS, FFP, FFP, FFP, FFP, 1 /*relu*/);
  wmma_gemm<<<cdiv((long long)wp * 32, 128), 128, 0, stream>>>(
      f1, Bp, bp, NLB, out, N_GRAPHS, FFP, NLP, NLB, NLB, 2 /*sigmoid*/);
}